// SupConLoss_9835475108099
// MI455X (gfx1250) — compile-verified
//
#include <hip/hip_runtime.h>
#include <hip/hip_bf16.h>
#include <math.h>

// SupCon loss, fully fused for gfx1250 (MI455X).
//   features: (2048, 2, 256) f32   cf row t -> features[(t%2048)][(t/2048)][:]
//   labels:   (2048,) int
//   out:      scalar f32 = contrastive loss + distance penalty
//
// Core: two N x N Grams (K=192 head, K=92 tail), N=4096, never materialized.
// Per 16x16 tile: V_WMMA_F32_16X16X4_F32 (exact f32 on the matrix pipe).
// j-tiles are streamed into LDS by the Tensor Data Mover (tensor_load_to_lds),
// double-buffered so DMA of tile jt+1 overlaps WMMA compute of tile jt.
// Row stats (online softmax max/sum, mask*logit sum, pos_count, mask*dist sum)
// kept as per-lane partials, combined with xor-shuffles at the end.

typedef __attribute__((ext_vector_type(2)))  float        v2f;
typedef __attribute__((ext_vector_type(8)))  float        v8f;
typedef __attribute__((ext_vector_type(4)))  unsigned int v4u;
typedef __attribute__((ext_vector_type(8)))  int          v8i;
typedef __attribute__((ext_vector_type(4)))  int          v4i;

#define NROWS 4096
#define DIM   256
#define SPLIT 192
#define TAIL  164
#define RS    260          // padded LDS row stride (floats): 260 % 64 == 4 -> conflict-free
#define IBLK  128          // i-rows per workgroup (8 waves x 16)
#define INV_T (1.0f/0.07f)

__device__ __forceinline__ size_t cfOffset(int t) {
    // cf row t lives at features[(t%2048)*512 + (t/2048)*256]
    return (size_t)(t & 2047) * 512 + (size_t)(t >> 11) * 256;
}

// ---- TDM: DMA one 16x256 f32 tile (row stride 512 floats) into LDS with
//      4-DWORD padding after every 256 DWORDs (-> 260-float LDS row stride).
__device__ __forceinline__ void tdm_load_tile(unsigned int lds_byte_addr,
                                              const float* gptr) {
    unsigned long long ga = (unsigned long long)(uintptr_t)gptr;
    v4u g0;
    g0.x = 1u;                                   // count=1, user descriptor
    g0.y = lds_byte_addr;                        // LDS dest (bytes)
    g0.z = (unsigned int)ga;                     // global_addr[31:0]
    g0.w = (unsigned int)(ga >> 32)              // global_addr[56:32]
           | (2u << 30);                         // type = 2 ("image")
    v8i g1;
    g1[0] = (2 << 16)                            // data_size = 4 bytes
          | (1 << 20)                            // pad_enable
          | (7 << 22)                            // pad_interval: every 256 DWORDs
          | (3 << 25);                           // pad_amount: 4 DWORDs
    g1[1] = (int)(256u << 16);                   // tensor_dim0[15:0] = 256
    g1[2] = (int)(16u  << 16);                   // tensor_dim0 hi=0; tensor_dim1 lo = 16
    g1[3] = (int)(256u << 16);                   // tensor_dim1 hi=0; tile_dim0 = 256
    g1[4] = 16;                                  // tile_dim1 = 16; tile_dim2 = 0
    g1[5] = 512;                                 // tensor_dim0_stride = 512 elements
    g1[6] = 0;
    g1[7] = 0;
    v4i z4 = {0, 0, 0, 0};                       // 2D tensor: groups 2/3 unused
    v8i z8 = {0, 0, 0, 0, 0, 0, 0, 0};
    // 6-arg toolchain variant: (g0 v4u, g1 v8i, g2 v4i, g3 v4i, v8i, cpol)
    __builtin_amdgcn_tensor_load_to_lds(g0, g1, z4, z4, z8, 0);
}

// ---- pass 0: tail squared norms + zero the output accumulator ----
__global__ void supcon_init(const float* __restrict__ feat,
                            float* __restrict__ sq,
                            float* __restrict__ out) {
    int t = blockIdx.x * blockDim.x + threadIdx.x;
    if (t == 0) out[0] = 0.0f;
    if (t < NROWS) {
        const float* row = feat + cfOffset(t);
        float s = 0.0f;
        #pragma unroll 4
        for (int k = TAIL; k < DIM; ++k) { float v = row[k]; s += v * v; }
        sq[t] = s;
    }
}

// ---- pass 1: fused dual-Gram + row reductions ----
__global__ void __launch_bounds__(256)
supcon_main(const float* __restrict__ feat,
            const int*   __restrict__ labels,
            const float* __restrict__ sq,
            float*       __restrict__ out) {
    extern __shared__ float smem[];
    float* sI  = smem;                       // IBLK x RS  (i-block)
    float* sJ0 = smem + IBLK * RS;           // 16 x RS    (j-tile buffer 0)
    float* sJ1 = sJ0 + 16 * RS;              // 16 x RS    (j-tile buffer 1)
    const unsigned int sJoff0 = (unsigned int)(uintptr_t)sJ0;  // low 32b of generic
    const unsigned int sJoff1 = (unsigned int)(uintptr_t)sJ1;  //  ptr = LDS address

    const int tid   = threadIdx.x;
    const int lane  = tid & 31;
    const int wave  = tid >> 5;
    const int iblk0 = blockIdx.x * IBLK;

    // Kick off DMA of j-tile 0 while we stage the i-block.
    if (wave == 0) tdm_load_tile(sJoff0, feat + cfOffset(0));

    // Stage the 128-row i-block into LDS with b128 loads.
    for (int t = tid; t < IBLK * (DIM / 4); t += 256) {
        int row = t >> 6, c4 = t & 63;
        float4 v = *(const float4*)(feat + cfOffset(iblk0 + row) + c4 * 4);
        *(float4*)(sI + row * RS + c4 * 4) = v;
    }

    const int ibase = iblk0 + wave * 16;
    const int half  = lane >> 4;            // 0: rows m=0..7, 1: rows m=8..15
    const int mbase = half * 8;

    // Per-lane row constants (same addr across half-wave -> cached)
    float sqI[8]; int labI[8];
    #pragma unroll
    for (int r = 0; r < 8; ++r) {
        int i = ibase + mbase + r;
        labI[r] = labels[i & 2047];
        sqI[r]  = sq[i];
    }

    // Per-lane partial row stats (columns j with j%16 == lane%16)
    float mx[8], ss[8], As[8], Pc[8], Ds[8];
    #pragma unroll
    for (int r = 0; r < 8; ++r) { mx[r] = -3.0e38f; ss[r] = 0.f; As[r] = 0.f; Pc[r] = 0.f; Ds[r] = 0.f; }

    // WMMA f32 16x16x4 operand addressing (A and B share the lane pattern):
    //   A: lane holds row (lane%16) of i-tile, k-pair (lane/16)*2 + {0,1}
    //   B: lane holds row (lane%16) of j-tile (B = Xj^T), same k-pair
    const float* aBase = sI + (wave * 16 + (lane & 15)) * RS + (half << 1);

    for (int jt = 0; jt < NROWS / 16; ++jt) {
        __syncthreads();  // iter jt-1 compute done -> buffer (jt+1)&1 is free
        if (wave == 0) {
            if (jt + 1 < NROWS / 16) {
                tdm_load_tile((jt & 1) ? sJoff0 : sJoff1,      // buffer (jt+1)&1
                              feat + cfOffset((jt + 1) * 16));
                __builtin_amdgcn_s_wait_tensorcnt(1);  // in-order TDM: tile jt landed
            } else {
                __builtin_amdgcn_s_wait_tensorcnt(0);
            }
        }
        __syncthreads();  // publish tile jt to all waves

        const float* sJ = (jt & 1) ? sJ1 : sJ0;
        const float* bBase = sJ + (lane & 15) * RS + (half << 1);

        v8f accF = {};   // head Gram (K=0..191)
        v8f accT = {};   // tail Gram (K=164..255)
        #pragma unroll
        for (int k = 0; k < SPLIT; k += 4) {
            v2f a = *(const v2f*)(aBase + k);
            v2f b = *(const v2f*)(bBase + k);
            accF = __builtin_amdgcn_wmma_f32_16x16x4_f32(false, a, false, b,
                                                         (short)0, accF, false, false);
        }
        #pragma unroll
        for (int k = TAIL; k < DIM; k += 4) {
            v2f a = *(const v2f*)(aBase + k);
            v2f b = *(const v2f*)(bBase + k);
            accT = __builtin_amdgcn_wmma_f32_16x16x4_f32(false, a, false, b,
                                                         (short)0, accT, false, false);
        }

        // Epilogue: this lane owns column n = lane%16 of the 16x16 tile,
        // rows m = mbase + r living in acc[r].
        const int n    = lane & 15;
        const int j    = jt * 16 + n;
        const int labJ = labels[j & 2047];
        const float sqJ = sq[j];
        #pragma unroll
        for (int r = 0; r < 8; ++r) {
            int   i     = ibase + mbase + r;
            float logit = accF[r] * INV_T;
            bool  diag  = (i == j);
            float msk   = (!diag && labI[r] == labJ) ? 1.0f : 0.0f;

            float nm = fmaxf(mx[r], logit);              // max includes diagonal
            float e  = diag ? 0.0f : __expf(logit - nm); // sum excludes diagonal
            ss[r] = ss[r] * __expf(mx[r] - nm) + e;
            mx[r] = nm;

            As[r] += msk * logit;
            Pc[r] += msk;

            float d2 = fmaxf(sqI[r] + sqJ - 2.0f * accT[r], 0.0f);
            float dd = (d2 > 1e-12f) ? sqrtf(d2) : 0.0f;
            Ds[r] += msk * dd;
        }
    }

    // Combine partials across the 16 lanes of each half-wave.
    #pragma unroll
    for (int r = 0; r < 8; ++r) {
        #pragma unroll
        for (int off = 1; off < 16; off <<= 1) {
            float om = __shfl_xor(mx[r], off, 32);
            float os = __shfl_xor(ss[r], off, 32);
            float nm = fmaxf(mx[r], om);
            ss[r] = ss[r] * __expf(mx[r] - nm) + os * __expf(om - nm);
            mx[r] = nm;
            As[r] += __shfl_xor(As[r], off, 32);
            Pc[r] += __shfl_xor(Pc[r], off, 32);
            Ds[r] += __shfl_xor(Ds[r], off, 32);
        }
    }

    if ((lane & 15) == 0) {   // lane 0: rows 0..7, lane 16: rows 8..15
        float total = 0.0f;
        #pragma unroll
        for (int r = 0; r < 8; ++r) {
            // mean_log_prob_pos = (A - P*max)/P - log(S)
            float mlpp = (As[r] - Pc[r] * mx[r]) / Pc[r] - __logf(ss[r]);
            total += -mlpp * (1.0f / (float)NROWS)      // contrastive term (T/BT = 1)
                     - 0.005f * Ds[r] / Pc[r];          // distance penalty (sum, not mean)
        }
        atomicAdd(out, total);
    }
}

extern "C" void kernel_launch(void* const* d_in, const int* in_sizes, int n_in,
                              void* d_out, int out_size, void* d_ws, size_t ws_size,
                              hipStream_t stream) {
    const float* feat   = (const float*)d_in[0];
    const int*   labels = (const int*)d_in[1];   // JAX default x64-off -> int32
    float*       out    = (float*)d_out;
    float*       sq     = (float*)d_ws;          // 4096 floats of tail sq norms

    supcon_init<<<(NROWS + 255) / 256, 256, 0, stream>>>(feat, sq, out);

    size_t shmem = (size_t)(IBLK * RS + 2 * 16 * RS) * sizeof(float);
    supcon_main<<<NROWS / IBLK, 256, shmem, stream>>>(feat, labels, sq, out);
}